// SGINClassification_84035330113568
// MI455X (gfx1250) — compile-verified
//
#include <hip/hip_runtime.h>
#include <hip/hip_bf16.h>
#include <math.h>

#define G_GRAPHS  1024
#define NLAYERS   4
#define NCLASSES  64

typedef float v2f __attribute__((ext_vector_type(2)));
typedef float v8f __attribute__((ext_vector_type(8)));

__device__ __forceinline__ void atomAddF(float* p, float v) {
    // guarantees native global_atomic_add_f32 (no CAS loop) on gfx1250
    unsafeAtomicAdd(p, v);
}

// ---- setup kernels -------------------------------------------------------

__global__ void k_init(float* __restrict__ deg, float* __restrict__ x,
                       float* __restrict__ s, float* __restrict__ counts, int n) {
    int i = blockIdx.x * blockDim.x + threadIdx.x;
    if (i < n) { deg[i] = 1.0f; x[i] = 1.0f; }          // deg starts at 1 (self loop)
    if (i < G_GRAPHS * NLAYERS) s[i] = 0.0f;
    if (i < G_GRAPHS) counts[i] = 0.0f;
}

__global__ void k_deg(const int* __restrict__ dst, float* __restrict__ deg, int e) {
    int i = blockIdx.x * blockDim.x + threadIdx.x;
    if (i < e) atomAddF(&deg[dst[i]], 1.0f);
}

__global__ void k_finalize(float* __restrict__ dinv, const int* __restrict__ batch,
                           float* __restrict__ counts, int n) {
    int i = blockIdx.x * blockDim.x + threadIdx.x;
    if (i < n) {
        float d = dinv[i];
        dinv[i] = (d > 0.0f) ? rsqrtf(d) : 0.0f;
        atomAddF(&counts[batch[i]], 1.0f);
    }
}

__global__ void k_norm(const int* __restrict__ src, const int* __restrict__ dst,
                       const float* __restrict__ dinv, float* __restrict__ norm, int e) {
    int i = blockIdx.x * blockDim.x + threadIdx.x;
    if (i < e) norm[i] = dinv[src[i]] * dinv[dst[i]];
}

// ---- one hop of D^{-1/2}(A+I)D^{-1/2} x ----------------------------------

__global__ void k_hop_init(const float* __restrict__ dinv, const float* __restrict__ x,
                           float* __restrict__ xn, int n) {
    int i = blockIdx.x * blockDim.x + threadIdx.x;
    if (i < n) { float d = dinv[i]; xn[i] = d * d * x[i]; }   // self-loop term
}

__global__ void k_hop_edges(const int* __restrict__ src, const int* __restrict__ dst,
                            const float* __restrict__ norm, const float* __restrict__ x,
                            float* __restrict__ xn, int e) {
    int i = blockIdx.x * blockDim.x + threadIdx.x;
    if (i < e) atomAddF(&xn[dst[i]], norm[i] * x[src[i]]);
}

__global__ void k_hop_edges_rc(const int* __restrict__ src, const int* __restrict__ dst,
                               const float* __restrict__ dinv, const float* __restrict__ x,
                               float* __restrict__ xn, int e) {
    int i = blockIdx.x * blockDim.x + threadIdx.x;
    if (i < e) {
        int s0 = src[i], d0 = dst[i];
        atomAddF(&xn[d0], dinv[s0] * dinv[d0] * x[s0]);
    }
}

// ---- per-graph sum of x (pooling), wave32 segmented pre-reduction --------

__global__ void k_reduce(const float* __restrict__ xn, const int* __restrict__ batch,
                         float* __restrict__ s_layer, int n) {
    int i = blockIdx.x * blockDim.x + threadIdx.x;
    int lane = threadIdx.x & 31;
    float v = 0.0f; int b = -1;
    if (i < n) { v = xn[i]; b = batch[i]; }
    // segmented suffix-sum over the wave (batch is sorted -> runs are contiguous)
    #pragma unroll
    for (int delta = 1; delta < 32; delta <<= 1) {
        float v2 = __shfl_down(v, delta, 32);
        int   b2 = __shfl_down(b, delta, 32);
        if ((lane + delta < 32) && (b2 == b)) v += v2;
    }
    int bprev = __shfl_up(b, 1, 32);
    bool head = (lane == 0) || (bprev != b);
    if (head && b >= 0) atomAddF(&s_layer[b], v);
}

// ---- finale: score = Z[1024x4] @ W^T[4x64] via V_WMMA_F32_16X16X4_F32,
// ---- fused bias + log_softmax. One wave per 16-graph tile. ---------------

__global__ void __launch_bounds__(32)
k_finale(const float* __restrict__ s, const float* __restrict__ counts,
         const float* __restrict__ W, const float* __restrict__ bvec,
         float* __restrict__ out) {
    __shared__ float lds_score[16 * NCLASSES];
    __shared__ float lds_bsum[NCLASSES];

    const int lane  = threadIdx.x;          // 0..31, EXEC all-ones before WMMA
    const int gbase = blockIdx.x * 16;
    const int M     = lane & 15;
    const int kb    = (lane >> 4) << 1;     // K pair: {0,1} or {2,3} (ISA 16x4 f32 A layout)
    const int g     = gbase + M;

    float cnt = counts[g];
    float ic  = (cnt > 0.0f) ? (1.0f / cnt) : 0.0f;

    // A (16x4): A[M][k] = s[k][g] / counts[g]
    v2f a;
    a.x = s[(kb + 0) * G_GRAPHS + g] * ic;
    a.y = s[(kb + 1) * G_GRAPHS + g] * ic;

    const int col    = lane & 15;
    const int rowoff = (lane >> 4) << 3;    // D rows: VGPR v -> M = v (lanes 0-15) / v+8 (16-31)

    #pragma unroll
    for (int t = 0; t < 4; ++t) {           // 4 class tiles of 16
        v2f bm;                              // B (4x16): B[k][n] = W[k][t*16+n]
        bm.x = W[(kb + 0) * NCLASSES + t * 16 + col];
        bm.y = W[(kb + 1) * NCLASSES + t * 16 + col];
        v8f c = {};
        c = __builtin_amdgcn_wmma_f32_16x16x4_f32(
                /*neg_a=*/false, a, /*neg_b=*/false, bm,
                /*c_mod=*/(short)0, c, /*reuse_a=*/false, /*reuse_b=*/false);
        #pragma unroll
        for (int v = 0; v < 8; ++v)
            lds_score[(rowoff + v) * NCLASSES + t * 16 + col] = c[v];
    }

    // Bsum[c] = sum_i b[i][c]
    lds_bsum[lane]      = bvec[0 * 64 + lane]      + bvec[1 * 64 + lane]
                        + bvec[2 * 64 + lane]      + bvec[3 * 64 + lane];
    lds_bsum[lane + 32] = bvec[0 * 64 + lane + 32] + bvec[1 * 64 + lane + 32]
                        + bvec[2 * 64 + lane + 32] + bvec[3 * 64 + lane + 32];
    __syncthreads();

    if (lane < 16) {
        int row = lane;
        int gg  = gbase + row;
        bool biased = counts[gg] > 0.0f;
        float m = -3.402823466e38f;
        #pragma unroll 4
        for (int c2 = 0; c2 < NCLASSES; ++c2) {
            float val = lds_score[row * NCLASSES + c2] + (biased ? lds_bsum[c2] : 0.0f);
            lds_score[row * NCLASSES + c2] = val;
            m = fmaxf(m, val);
        }
        float ssum = 0.0f;
        #pragma unroll 4
        for (int c2 = 0; c2 < NCLASSES; ++c2)
            ssum += __expf(lds_score[row * NCLASSES + c2] - m);
        float lg = m + __logf(ssum);
        #pragma unroll 4
        for (int c2 = 0; c2 < NCLASSES; ++c2)
            out[gg * NCLASSES + c2] = lds_score[row * NCLASSES + c2] - lg;
    }
}

// ---- host launch ---------------------------------------------------------

extern "C" void kernel_launch(void* const* d_in, const int* in_sizes, int n_in,
                              void* d_out, int out_size, void* d_ws, size_t ws_size,
                              hipStream_t stream) {
    const int*   edge_index = (const int*)d_in[0];   // (2, E)
    const int*   batch      = (const int*)d_in[1];   // (N,)
    const float* W          = (const float*)d_in[2]; // (4, 1, 64)
    const float* bvec       = (const float*)d_in[3]; // (4, 64)
    float*       out        = (float*)d_out;         // (1024, 64)

    const int E = in_sizes[0] / 2;
    const int N = in_sizes[1];
    const int* src = edge_index;
    const int* dst = edge_index + E;

    char* ws = (char*)d_ws;
    size_t off = 0;
    auto wsalloc = [&](size_t bytes) -> void* {
        void* p = (void*)(ws + off);
        off = (off + bytes + 255) & ~(size_t)255;
        return p;
    };
    float* dinv   = (float*)wsalloc((size_t)N * 4);            // deg, then rsqrt in place
    float* xa     = (float*)wsalloc((size_t)N * 4);
    float* xb     = (float*)wsalloc((size_t)N * 4);
    float* svec   = (float*)wsalloc((size_t)G_GRAPHS * NLAYERS * 4);
    float* counts = (float*)wsalloc((size_t)G_GRAPHS * 4);
    float* norm   = nullptr;
    if (off + (size_t)E * 4 <= ws_size)                        // 26 MB: use if ws allows
        norm = (float*)wsalloc((size_t)E * 4);

    const int nb = (N + 255) / 256;
    const int eb = (E + 255) / 256;

    k_init<<<nb, 256, 0, stream>>>(dinv, xa, svec, counts, N);
    k_deg<<<eb, 256, 0, stream>>>(dst, dinv, E);
    k_finalize<<<nb, 256, 0, stream>>>(dinv, batch, counts, N);
    if (norm) k_norm<<<eb, 256, 0, stream>>>(src, dst, dinv, norm, E);

    for (int i = 0; i < NLAYERS; ++i) {
        k_hop_init<<<nb, 256, 0, stream>>>(dinv, xa, xb, N);
        if (norm) k_hop_edges<<<eb, 256, 0, stream>>>(src, dst, norm, xa, xb, E);
        else      k_hop_edges_rc<<<eb, 256, 0, stream>>>(src, dst, dinv, xa, xb, E);
        k_reduce<<<nb, 256, 0, stream>>>(xb, batch, svec + (size_t)i * G_GRAPHS, N);
        float* t = xa; xa = xb; xb = t;
    }

    k_finale<<<G_GRAPHS / 16, 32, 0, stream>>>(svec, counts, W, bvec, out);
}